// Pooler_38792144617925
// MI455X (gfx1250) — compile-verified
//
#include <hip/hip_runtime.h>
#include <stdint.h>

#define OUTP   7
#define CTILE  32          // channels staged per block
#define MAXEL  1024        // max region elements per channel staged in LDS
#define NCH    256
#define NLVL_SHIFT 3       // NCH/CTILE == 8 blocks per roi

typedef unsigned int u32x4 __attribute__((ext_vector_type(4)));
typedef int          i32x8 __attribute__((ext_vector_type(8)));
typedef int          i32x4 __attribute__((ext_vector_type(4)));

__device__ __forceinline__ void edge_clamp(float c, int limit, int& lo, float& fr) {
    float cc = fminf(fmaxf(c, 0.0f), (float)(limit - 1));
    int l = (int)cc;                 // floor (cc >= 0)
    if (l > limit - 2) l = limit - 2;
    lo = l;
    fr = cc - (float)l;
}

__global__ __launch_bounds__(256)
void pooler_kernel(const float* __restrict__ f0, const float* __restrict__ f1,
                   const float* __restrict__ f2, const float* __restrict__ f3,
                   const float* __restrict__ rois, float* __restrict__ out, int N)
{
    __shared__ float smem[CTILE * MAXEL];   // 128 KB LDS tile (CDNA5: 320 KB/WGP)

    const int blk = blockIdx.x;
    const int n   = blk >> NLVL_SHIFT;
    const int ct  = blk & ((1 << NLVL_SHIFT) - 1);
    const int c0  = ct * CTILE;
    if (n >= N) return;

    const float* r = rois + (size_t)n * 5;
    const int   b   = (int)r[0];
    const float rx1 = r[1], ry1 = r[2], rx2 = r[3], ry2 = r[4];

    // ---- level selection (matches reference) ----
    float area = (rx2 - rx1 + 1.0f) * (ry2 - ry1 + 1.0f);
    float s    = sqrtf(area);
    float tl   = floorf(4.0f + log2f(s * (1.0f / 224.0f) + 1e-6f));
    tl = fminf(fmaxf(tl, 2.0f), 5.0f);
    const int lvl = (int)tl - 2;

    const float* fp; int H, W; float scale;
    switch (lvl) {
        case 0:  fp = f0; H = 200; W = 304; scale = 0.25f;    break;
        case 1:  fp = f1; H = 100; W = 152; scale = 0.125f;   break;
        case 2:  fp = f2; H = 50;  W = 76;  scale = 0.0625f;  break;
        default: fp = f3; H = 25;  W = 38;  scale = 0.03125f; break;
    }

    const float x1s = rx1 * scale, y1s = ry1 * scale;
    const float rw  = fmaxf(rx2 * scale - x1s, 1.0f);
    const float rh  = fmaxf(ry2 * scale - y1s, 1.0f);
    const float bw  = rw * (1.0f / OUTP), bh = rh * (1.0f / OUTP);

    // ---- region bounds: samples are monotonic, so first/last sample suffice ----
    // sample offsets: (sidx + 0.5)/2, sidx in [0,13] -> 0.25 .. 6.75
    int xl0, xl1, yl0, yl1; float dum;
    edge_clamp(x1s + bw * 0.25f, W, xl0, dum);
    edge_clamp(x1s + bw * 6.75f, W, xl1, dum);
    edge_clamp(y1s + bh * 0.25f, H, yl0, dum);
    edge_clamp(y1s + bh * 6.75f, H, yl1, dum);
    const int x0   = xl0, y0 = yl0;
    const int regW = xl1 - xl0 + 2;
    const int regH = yl1 - yl0 + 2;
    const int elems = regW * regH;
    const bool useLds = (elems <= MAXEL);

    const size_t outBase = (size_t)n * (size_t)(NCH * OUTP * OUTP) + (size_t)c0 * (OUTP * OUTP);

    if (useLds) {
        // ---- stage [CTILE][regH][regW] rectangle into LDS via the Tensor Data Mover ----
        // Workgroup-relative LDS byte offset of smem: low 32 bits of its generic address
        // (generic LDS addresses are {SHARED_BASE, lds_offset}).  Feeding this through the
        // descriptor also escapes the pointer so the ds_loads below are not assumed undef.
        const unsigned ldsOff = (unsigned)(uintptr_t)(void*)&smem[0];
#if defined(__has_builtin) && __has_builtin(__builtin_amdgcn_tensor_load_to_lds)
        if (threadIdx.x < 32) {
            const uint64_t ga = (uint64_t)(uintptr_t)fp +
                4ull * ((uint64_t)(b * NCH + c0) * (uint64_t)(H * W)
                        + (uint64_t)y0 * (uint64_t)W + (uint64_t)x0);
            u32x4 g0; i32x8 g1; i32x4 g2, g3; i32x8 g4;
            // D# group 0: count=1, lds_addr, global addr, type=2
            g0[0] = 1u;
            g0[1] = ldsOff;
            g0[2] = (unsigned)(ga & 0xffffffffu);
            g0[3] = (unsigned)((ga >> 32) & 0x01ffffffu) | (2u << 30);
            // D# group 1: data_size=4B; dims/tiles/strides
            const uint64_t st0 = (uint64_t)W;           // row (y) stride (elements)
            const uint64_t st1 = (uint64_t)(H * W);     // channel stride (elements)
            const unsigned td0 = (unsigned)(W - x0);    // remaining tensor extent from tile start
            const unsigned td1 = (unsigned)(H - y0);
            g1[0] = (int)(2u << 16);                                        // data_size=2 (4B)
            g1[1] = (int)((td0 & 0xffffu) << 16);                           // tensor_dim0 lo
            g1[2] = (int)(((td0 >> 16) & 0xffffu) | ((td1 & 0xffffu) << 16));
            g1[3] = (int)(((td1 >> 16) & 0xffffu) | ((unsigned)regW << 16)); // tile_dim0
            g1[4] = (int)(((unsigned)regH & 0xffffu) | ((unsigned)CTILE << 16)); // tile_dim1, tile_dim2
            g1[5] = (int)(unsigned)(st0 & 0xffffffffu);
            g1[6] = (int)((unsigned)((st0 >> 32) & 0xffffu) | ((unsigned)(st1 & 0xffffu) << 16));
            g1[7] = (int)(unsigned)(st1 >> 16);
            // D# group 2/3: tensor_dim2 = remaining channels; rest unused
            g2[0] = (int)(unsigned)(NCH - c0);
            g2[1] = 0; g2[2] = 0; g2[3] = 0;
            g3[0] = 0; g3[1] = 0; g3[2] = 0; g3[3] = 0;
            // 5th operand (clang-23 6-arg form): zero-filled
            g4[0] = 0; g4[1] = 0; g4[2] = 0; g4[3] = 0;
            g4[4] = 0; g4[5] = 0; g4[6] = 0; g4[7] = 0;
            __builtin_amdgcn_tensor_load_to_lds(g0, g1, g2, g3, g4, 0);
#if __has_builtin(__builtin_amdgcn_s_wait_tensorcnt)
            __builtin_amdgcn_s_wait_tensorcnt(0);
#else
            asm volatile("s_wait_tensorcnt 0x0" ::: "memory");
#endif
        }
#else
        // Fallback (toolchain without the TDM builtin): cooperative staged copy
        (void)ldsOff;
        for (int i = threadIdx.x; i < elems * CTILE; i += 256) {
            int cch = i / elems, rem = i - cch * elems;
            int yy = rem / regW, xx = rem - yy * regW;
            smem[i] = fp[(size_t)(b * NCH + c0 + cch) * (size_t)(H * W)
                         + (size_t)(y0 + yy) * W + (x0 + xx)];
        }
#endif
        asm volatile("" ::: "memory");
        __syncthreads();

        // ---- hot path: 16-tap bilinear entirely from LDS (ds_load) ----
        for (int idx = threadIdx.x; idx < CTILE * OUTP * OUTP; idx += 256) {
            const int cl = idx / (OUTP * OUTP);
            const int p  = idx - cl * (OUTP * OUTP);
            const int py = p / OUTP;
            const int px = p - py * OUTP;
            const int cbase = cl * regH;

            float acc = 0.0f;
#pragma unroll
            for (int dy = 0; dy < 2; ++dy) {
                const float yc = y1s + bh * ((float)(2 * py + dy) * 0.5f + 0.25f);
                const bool  vy = (yc >= -1.0f) && (yc <= (float)H);
                int yl; float fy; edge_clamp(yc, H, yl, fy);
#pragma unroll
                for (int dx = 0; dx < 2; ++dx) {
                    const float xc = x1s + bw * ((float)(2 * px + dx) * 0.5f + 0.25f);
                    const bool  vx = (xc >= -1.0f) && (xc <= (float)W);
                    if (vy && vx) {
                        int xl; float fx; edge_clamp(xc, W, xl, fx);
                        const int base = (cbase + (yl - y0)) * regW + (xl - x0);
                        const float v00 = smem[base];
                        const float v01 = smem[base + 1];
                        const float v10 = smem[base + regW];
                        const float v11 = smem[base + regW + 1];
                        const float top = v00 + fx * (v01 - v00);
                        const float bot = v10 + fx * (v11 - v10);
                        acc += top + fy * (bot - top);
                    }
                }
            }
            out[outBase + idx] = acc * 0.25f;
        }
    } else {
        // ---- rare oversized-region fallback: direct global gathers ----
        for (int idx = threadIdx.x; idx < CTILE * OUTP * OUTP; idx += 256) {
            const int cl = idx / (OUTP * OUTP);
            const int p  = idx - cl * (OUTP * OUTP);
            const int py = p / OUTP;
            const int px = p - py * OUTP;
            const float* cplane = fp + (size_t)(b * NCH + c0 + cl) * (size_t)(H * W);

            float acc = 0.0f;
#pragma unroll
            for (int dy = 0; dy < 2; ++dy) {
                const float yc = y1s + bh * ((float)(2 * py + dy) * 0.5f + 0.25f);
                const bool  vy = (yc >= -1.0f) && (yc <= (float)H);
                int yl; float fy; edge_clamp(yc, H, yl, fy);
#pragma unroll
                for (int dx = 0; dx < 2; ++dx) {
                    const float xc = x1s + bw * ((float)(2 * px + dx) * 0.5f + 0.25f);
                    const bool  vx = (xc >= -1.0f) && (xc <= (float)W);
                    if (vy && vx) {
                        int xl; float fx; edge_clamp(xc, W, xl, fx);
                        const float* q = cplane + (size_t)yl * W + xl;
                        const float v00 = q[0], v01 = q[1];
                        const float v10 = q[W], v11 = q[W + 1];
                        const float top = v00 + fx * (v01 - v00);
                        const float bot = v10 + fx * (v11 - v10);
                        acc += top + fy * (bot - top);
                    }
                }
            }
            out[outBase + idx] = acc * 0.25f;
        }
    }
}

extern "C" void kernel_launch(void* const* d_in, const int* in_sizes, int n_in,
                              void* d_out, int out_size, void* d_ws, size_t ws_size,
                              hipStream_t stream) {
    const float* f0   = (const float*)d_in[0];
    const float* f1   = (const float*)d_in[1];
    const float* f2   = (const float*)d_in[2];
    const float* f3   = (const float*)d_in[3];
    const float* rois = (const float*)d_in[4];
    float* out = (float*)d_out;

    const int N = in_sizes[4] / 5;                 // 1024 rois
    const dim3 grid((unsigned)(N * (NCH / CTILE))); // 8192 blocks
    pooler_kernel<<<grid, 256, 0, stream>>>(f0, f1, f2, f3, rois, out, N);
}